// RNN_7344394076603
// MI455X (gfx1250) — compile-verified
//
#include <hip/hip_runtime.h>
#include <math.h>

typedef __attribute__((ext_vector_type(2))) float v2f;
typedef __attribute__((ext_vector_type(8))) float v8f;

#define T_STEPS 512
#define BATCH   256
#define INSZ    128
#define NRN     512
#define KTOT    (NRN + INSZ)      // 640

#define NBLOCKS  8                // one block per 64-column N-slab
#define NTHREADS 512              // 16 wave32 -> one wave per 16-row M-tile
#define KPAD     644              // 640 + 4 dword pad: bank-conflict-free ds_load_b64
#define LDS_BYTES (64 * KPAD * 4) // 164,864 B  (<= 320 KB WGP LDS)

// Zero h-buffer-A (aliases d_out) and the barrier counter.
__global__ void rnn_init_kernel(float* __restrict__ hA, unsigned* __restrict__ counter) {
    int idx = blockIdx.x * blockDim.x + threadIdx.x;
    int stride = gridDim.x * blockDim.x;
    for (int i = idx; i < BATCH * NRN; i += stride) hA[i] = 0.0f;
    if (idx == 0) *counter = 0u;
}

__global__ void __launch_bounds__(NTHREADS, 1)
rnn_persistent_kernel(const float* __restrict__ x,    // [T, B, I]
                      const float* __restrict__ w1,   // [I, N]
                      const float* __restrict__ w2,   // [N, N]
                      const float* __restrict__ bias, // [1, N]
                      float* __restrict__ hA,         // aliases d_out
                      float* __restrict__ hB,         // in d_ws
                      unsigned* __restrict__ counter) // in d_ws
{
    extern __shared__ float ldsW[];      // [64][KPAD] : transposed [n][k] weight strip

    const int ns    = blockIdx.x;              // N-slab 0..7 (64 cols)
    const int mt    = threadIdx.x >> 5;        // M-tile 0..15 (wave id)
    const int lane  = threadIdx.x & 31;
    const int l16   = lane & 15;
    const int lhalf = lane >> 4;               // 0 or 1

    const int mrow = mt * 16 + l16;            // batch row this lane supplies for A
    const int ncb  = ns * 64 + l16;            // base global output column

    // ---- Stage [W2;W1] 64-column strip into LDS, transposed to [n][k] ----
    // Consecutive threads touch consecutive n for a fixed k -> coalesced global reads.
    for (int i = threadIdx.x; i < 64 * KTOT; i += NTHREADS) {
        const int k  = i >> 6;                 // 0..639
        const int nl = i & 63;                 // 0..63
        const int col = ns * 64 + nl;
        const float v = (k < NRN) ? w2[(size_t)k * NRN + col]
                                  : w1[(size_t)(k - NRN) * NRN + col];
        ldsW[nl * KPAD + k] = v;
    }
    __syncthreads();

    // Per-lane, per-tile LDS base: B[k'][n] lives at ldsW[n*KPAD + k'],
    // k' = k + lhalf*2 + {0,1} contiguous -> single ds_load_b64.
    const float* bb[4];
#pragma unroll
    for (int j = 0; j < 4; ++j) bb[j] = ldsW + (l16 + 16 * j) * KPAD + lhalf * 2;

    float biasv[4];
#pragma unroll
    for (int j = 0; j < 4; ++j) biasv[j] = bias[ncb + j * 16];

    for (int t = 0; t < T_STEPS; ++t) {
        const float* __restrict__ hsrc = (t & 1) ? hB : hA;
        float*       __restrict__ hdst = (t & 1) ? hA : hB; // t=511 (odd) -> hA = d_out
        const float* __restrict__ xt   = x + (size_t)t * (BATCH * INSZ);

        // Pull next timestep's input row ahead of use (only HBM-resident stream).
        if (t + 1 < T_STEPS) {
            const float* xn = xt + BATCH * INSZ + (size_t)mrow * INSZ;
            __builtin_prefetch(xn, 0, 0);
            __builtin_prefetch(xn + 64, 0, 0);
        }

        v8f acc[4] = {};   // 4 16x16 f32 accumulators (M=16, N=64 slab)

        // ---- Phase 1: h_prev @ W2  (K = 512), B from LDS ----
        const float* aptr1 = hsrc + (size_t)mrow * NRN + lhalf * 2;
#pragma unroll 2
        for (int k = 0; k < NRN; k += 4) {
            v2f a = *(const v2f*)(aptr1 + k);            // global b64, contiguous K
#pragma unroll
            for (int j = 0; j < 4; ++j) {
                v2f b = *(const v2f*)(bb[j] + k);        // ds_load_b64, bank-clean
                acc[j] = __builtin_amdgcn_wmma_f32_16x16x4_f32(
                    false, a, false, b, (short)0, acc[j], false, false);
            }
        }

        // ---- Phase 2: x_t @ W1  (K = 128), B from LDS at K-offset 512 ----
        const float* aptr2 = xt + (size_t)mrow * INSZ + lhalf * 2;
#pragma unroll 2
        for (int k = 0; k < INSZ; k += 4) {
            v2f a = *(const v2f*)(aptr2 + k);
#pragma unroll
            for (int j = 0; j < 4; ++j) {
                v2f b = *(const v2f*)(bb[j] + NRN + k);
                acc[j] = __builtin_amdgcn_wmma_f32_16x16x4_f32(
                    false, a, false, b, (short)0, acc[j], false, false);
            }
        }

        // ---- Epilogue: h = tanh(acc + bias) ----
        // C/D layout: lanes 0-15 -> N=lane, M=r; lanes 16-31 -> N=lane-16, M=r+8.
#pragma unroll
        for (int j = 0; j < 4; ++j) {
            const int n = ncb + j * 16;
#pragma unroll
            for (int r = 0; r < 8; ++r) {
                const int m = mt * 16 + r + lhalf * 8;
                hdst[(size_t)m * NRN + n] = tanhf(acc[j][r] + biasv[j]);
            }
        }

        // ---- Device-wide split barrier between timesteps (8 blocks) ----
        __threadfence();
        __syncthreads();
        if (threadIdx.x == 0) {
            const unsigned target = (unsigned)(t + 1) * NBLOCKS;
            atomicAdd(counter, 1u);
            while (__hip_atomic_load(counter, __ATOMIC_ACQUIRE,
                                     __HIP_MEMORY_SCOPE_AGENT) < target) {
                __builtin_amdgcn_s_sleep(1);
            }
        }
        __syncthreads();
    }
}

extern "C" void kernel_launch(void* const* d_in, const int* in_sizes, int n_in,
                              void* d_out, int out_size, void* d_ws, size_t ws_size,
                              hipStream_t stream) {
    const float* x    = (const float*)d_in[0]; // [512,256,128]
    const float* w1   = (const float*)d_in[1]; // [128,512]
    const float* w2   = (const float*)d_in[2]; // [512,512]
    const float* bias = (const float*)d_in[3]; // [1,512]

    float*    hA      = (float*)d_out;                 // h buffer A aliases output
    unsigned* counter = (unsigned*)d_ws;               // barrier counter
    float*    hB      = (float*)((char*)d_ws + 256);   // h buffer B (512 KB)

    // Raise dynamic-LDS cap to our 161 KB strip (non-stream call; capture-safe).
    (void)hipFuncSetAttribute((const void*)rnn_persistent_kernel,
                              hipFuncAttributeMaxDynamicSharedMemorySize, LDS_BYTES);

    rnn_init_kernel<<<128, 256, 0, stream>>>(hA, counter);
    rnn_persistent_kernel<<<NBLOCKS, NTHREADS, LDS_BYTES, stream>>>(
        x, w1, w2, bias, hA, hB, counter);
}